// EMDInformedInSARModel_85779086835984
// MI455X (gfx1250) — compile-verified
//
#include <hip/hip_runtime.h>
#include <hip/hip_bf16.h>

// ---------------------------------------------------------------------------
// signals = U (N x 12) x V (12 x Tpad), computed with V_WMMA_F32_16X16X4_F32.
//   U[n] = [off, trend, a_i*cos(p_i), a_i*sin(p_i) (i=0..3), 0, 0]
//   V[:,t] = [1, t, sin(w_i t), cos(w_i t) (i=0..3), 0, 0]
// so dot(U[n], V[:,t]) = off + trend*t + sum_i a_i*sin(w_i t + p_i).
// Store-bandwidth bound: 200 MB out @ 23.3 TB/s ~ 8.6 us floor.
// ---------------------------------------------------------------------------

typedef __attribute__((ext_vector_type(2))) float v2f;
typedef __attribute__((ext_vector_type(8))) float v8f;

#define TWO_PI_F 6.28318530717958647692f
#define SMOOTH_AMP 0.15f
#define SMOOTH_PH  0.10f

// ---------------------------------------------------------------------------
// Kernel 1: cos/sin of every seasonal phase (N*4 values).
// ---------------------------------------------------------------------------
__global__ void trig_kernel(const float* __restrict__ ph,
                            float* __restrict__ cosph,
                            float* __restrict__ sinph, int n) {
  int i = blockIdx.x * blockDim.x + threadIdx.x;
  if (i < n) {
    float s, c;
    __sincosf(ph[i], &s, &c);
    cosph[i] = c;
    sinph[i] = s;
  }
}

// ---------------------------------------------------------------------------
// Kernel 2: build U (Npad x 12, row-major). One thread per row.
// ---------------------------------------------------------------------------
__global__ void build_u_kernel(const float* __restrict__ off,
                               const float* __restrict__ trend,
                               const float* __restrict__ amps,   // [N][4]
                               const float* __restrict__ phs,    // [N][4]
                               const float* __restrict__ w,      // [N][8]
                               const float* __restrict__ bmin,   // [4][N]
                               const float* __restrict__ bmax,   // [4][N]
                               const int*   __restrict__ nbr,    // [N][8]
                               const float* __restrict__ cosph,  // [N][4]
                               const float* __restrict__ sinph,  // [N][4]
                               float* __restrict__ U, int N, int Npad) {
  int n = blockIdx.x * blockDim.x + threadIdx.x;
  if (n >= Npad) return;
  float* urow = U + (size_t)n * 12;
  if (n >= N) {  // zero pad rows so WMMA on tail tiles is harmless
    #pragma unroll
    for (int q = 0; q < 12; ++q) urow[q] = 0.0f;
    return;
  }
  float u[12];
  u[0] = off[n];
  u[1] = trend[n];
  u[10] = 0.0f;
  u[11] = 0.0f;

  int   idx[8];
  float wt[8];
  const int base = n * 8;
  #pragma unroll
  for (int k = 0; k < 8; ++k) { idx[k] = nbr[base + k]; wt[k] = w[base + k]; }

  #pragma unroll
  for (int i = 0; i < 4; ++i) {
    float amp = amps[n * 4 + i];
    float ph  = phs [n * 4 + i];
    float na = 0.0f, ar = 0.0f, ai = 0.0f;
    #pragma unroll
    for (int k = 0; k < 8; ++k) {
      int j = idx[k] * 4 + i;
      na = fmaf(amps [j], wt[k], na);
      ar = fmaf(cosph[j], wt[k], ar);
      ai = fmaf(sinph[j], wt[k], ai);
    }
    float samp = (1.0f - SMOOTH_AMP) * amp + SMOOTH_AMP * na;
    samp = fminf(fmaxf(samp, bmin[i * N + n]), bmax[i * N + n]);

    float ci, cr;
    __sincosf(ph, &ci, &cr);                 // ci = sin, cr = cos
    float mr = (1.0f - SMOOTH_PH) * cr + SMOOTH_PH * ar;
    float mi = (1.0f - SMOOTH_PH) * ci + SMOOTH_PH * ai;
    // cos/sin of atan2(mi, mr) without trig:
    float inv = rsqrtf(fmaxf(mr * mr + mi * mi, 1e-30f));
    u[2 + 2 * i] = samp * mr * inv;          // pairs with sin(w t)
    u[3 + 2 * i] = samp * mi * inv;          // pairs with cos(w t)
  }
  #pragma unroll
  for (int q = 0; q < 12; ++q) urow[q] = u[q];
}

// ---------------------------------------------------------------------------
// Kernel 3: build V packed as Vp[pair][t][2] (pair = k/2, 6 pairs, Tpad cols).
//   pair 0: (1, t) ; pair 1+i: (sin(w_i t), cos(w_i t)) ; pair 5: (0, 0)
// ---------------------------------------------------------------------------
__global__ void build_v_kernel(const float* __restrict__ tv,
                               float* __restrict__ Vp, int T, int TP) {
  int t = blockIdx.x * blockDim.x + threadIdx.x;
  if (t >= TP) return;
  const bool valid = (t < T);
  float tt = valid ? tv[t] : 0.0f;
  Vp[((size_t)0 * TP + t) * 2 + 0] = valid ? 1.0f : 0.0f;
  Vp[((size_t)0 * TP + t) * 2 + 1] = tt;
  const float freqs[4] = {4.0f, 2.0f, 1.0f, 0.5f};  // 1/period
  #pragma unroll
  for (int i = 0; i < 4; ++i) {
    float s, c;
    __sincosf(TWO_PI_F * freqs[i] * tt, &s, &c);
    if (!valid) { s = 0.0f; c = 0.0f; }
    Vp[((size_t)(1 + i) * TP + t) * 2 + 0] = s;
    Vp[((size_t)(1 + i) * TP + t) * 2 + 1] = c;
  }
  Vp[((size_t)5 * TP + t) * 2 + 0] = 0.0f;   // K=10,11 zero pad
  Vp[((size_t)5 * TP + t) * 2 + 1] = 0.0f;
}

// ---------------------------------------------------------------------------
// Kernel 4: the GEMM. One wave per 16-row n-tile; A fragment loaded once and
// reused across all t-tiles; 3 chained V_WMMA_F32_16X16X4_F32 per 16x16 tile.
// Fast path (all rows valid, all cols valid) does 8 unguarded b32 row stores;
// the single partial t-tile and the (nonexistent here) partial n-tile are
// peeled so the hot loop has no exec-mask traffic.
// ---------------------------------------------------------------------------
__device__ __forceinline__ v8f tile_mma(v2f a0, v2f a1, v2f a2,
                                        const v2f* pb0, const v2f* pb1,
                                        const v2f* pb2, int col) {
  const v2f b0 = pb0[col];
  const v2f b1 = pb1[col];
  const v2f b2 = pb2[col];
  v8f acc = {};
  acc = __builtin_amdgcn_wmma_f32_16x16x4_f32(false, a0, false, b0,
                                              (short)0, acc, false, false);
  acc = __builtin_amdgcn_wmma_f32_16x16x4_f32(false, a1, false, b1,
                                              (short)0, acc, false, false);
  acc = __builtin_amdgcn_wmma_f32_16x16x4_f32(false, a2, false, b2,
                                              (short)0, acc, false, false);
  return acc;
}

__global__ __launch_bounds__(256)
void signal_wmma_kernel(const float* __restrict__ U,   // [Npad][12]
                        const float* __restrict__ Vp,  // [6][TP][2]
                        float* __restrict__ out,       // [N][T]
                        int N, int T, int TP, int nTiles) {
  const int wave = blockIdx.x * (blockDim.x >> 5) + (threadIdx.x >> 5);
  const int lane = threadIdx.x & 31;
  // Wave-uniform launch guard as a scalar branch (EXEC untouched for WMMA).
  if (__builtin_amdgcn_readfirstlane(wave) >= nTiles) return;

  const int n0   = wave << 4;
  const int half = lane >> 4;           // 0 or 1
  const int l16  = lane & 15;

  // A fragments: U[n0 + l16][kb*4 + 2*half + {0,1}]
  const float* urow = U + (size_t)(n0 + l16) * 12 + 2 * half;
  const v2f a0 = *(const v2f*)(urow + 0);
  const v2f a1 = *(const v2f*)(urow + 4);
  const v2f a2 = *(const v2f*)(urow + 8);

  // B base pointers for this lane (pair p = kb*2 + half, column l16).
  const v2f* pb0 = (const v2f*)(Vp + ((size_t)(0 + half) * TP + l16) * 2);
  const v2f* pb1 = (const v2f*)(Vp + ((size_t)(2 + half) * TP + l16) * 2);
  const v2f* pb2 = (const v2f*)(Vp + ((size_t)(4 + half) * TP + l16) * 2);

  const int rbase = n0 + 8 * half;      // first of this lane-half's 8 rows
  float* orow = out + (size_t)rbase * T + l16;

  const int fullT  = T >> 4;            // t-tiles with all 16 cols in range
  const int tTiles = TP >> 4;
  const int rowsFull =
      __builtin_amdgcn_readfirstlane((int)(n0 + 16 <= N));  // wave-uniform

  if (rowsFull) {
    int jt = 0;
    for (; jt < fullT; ++jt) {          // hot loop: no guards at all
      const v8f acc = tile_mma(a0, a1, a2, pb0, pb1, pb2, jt << 4);
      float* o = orow + (jt << 4);
      #pragma unroll
      for (int r = 0; r < 8; ++r) o[(size_t)r * T] = acc[r];
    }
    for (; jt < tTiles; ++jt) {         // peeled partial-column tile
      const v8f acc = tile_mma(a0, a1, a2, pb0, pb1, pb2, jt << 4);
      if ((jt << 4) + l16 < T) {
        float* o = orow + (jt << 4);
        #pragma unroll
        for (int r = 0; r < 8; ++r) o[(size_t)r * T] = acc[r];
      }
    }
  } else {
    // Rare tail n-tile (N % 16 != 0): fully guarded stores.
    for (int jt = 0; jt < tTiles; ++jt) {
      const v8f acc = tile_mma(a0, a1, a2, pb0, pb1, pb2, jt << 4);
      const int c = (jt << 4) + l16;
      if (c < T) {
        #pragma unroll
        for (int r = 0; r < 8; ++r) {
          if (rbase + r < N) out[(size_t)(rbase + r) * T + c] = acc[r];
        }
      }
    }
  }
}

// ---------------------------------------------------------------------------
// Launch
// ---------------------------------------------------------------------------
extern "C" void kernel_launch(void* const* d_in, const int* in_sizes, int n_in,
                              void* d_out, int out_size, void* d_ws, size_t ws_size,
                              hipStream_t stream) {
  const float* time_vec = (const float*)d_in[0];
  const float* off      = (const float*)d_in[1];
  const float* trend    = (const float*)d_in[2];
  const float* amps     = (const float*)d_in[3];
  const float* phs      = (const float*)d_in[4];
  const float* w        = (const float*)d_in[5];
  const float* bmin     = (const float*)d_in[6];
  const float* bmax     = (const float*)d_in[7];
  const int*   nbr      = (const int*)  d_in[8];
  float* out = (float*)d_out;

  const int T    = in_sizes[0];            // 1000
  const int N    = in_sizes[1];            // 50000
  const int TP   = ((T + 15) / 16) * 16;   // 1008
  const int Npad = ((N + 15) / 16) * 16;   // 50000 (already multiple of 16)
  const int nTiles = Npad / 16;

  // Workspace carve-up (floats):  U | Vp | cosph | sinph  (~4.1 MB total)
  float* ws    = (float*)d_ws;
  float* U     = ws;
  float* Vp    = U  + (size_t)Npad * 12;
  float* cosph = Vp + (size_t)12 * TP;
  float* sinph = cosph + (size_t)N * 4;

  const int B = 256;
  trig_kernel<<<(N * 4 + B - 1) / B, B, 0, stream>>>(phs, cosph, sinph, N * 4);
  build_u_kernel<<<(Npad + B - 1) / B, B, 0, stream>>>(
      off, trend, amps, phs, w, bmin, bmax, nbr, cosph, sinph, U, N, Npad);
  build_v_kernel<<<(TP + B - 1) / B, B, 0, stream>>>(time_vec, Vp, T, TP);

  const int wavesPerBlock = B / 32;
  const int grid = (nTiles + wavesPerBlock - 1) / wavesPerBlock;
  signal_wmma_kernel<<<grid, B, 0, stream>>>(U, Vp, out, N, T, TP, nTiles);
}